// VGAE_Encoder_72189810312082
// MI455X (gfx1250) — compile-verified
//
#include <hip/hip_runtime.h>

// VGAE encoder for MI455X (gfx1250, wave32).
// GEMMs use V_WMMA_F32_16X16X4_F32 (full fp32 — op is bandwidth bound, so no
// reason to quantize). Edge aggregation uses wave-per-edge gather + f32 atomics.

#define NN   50000
#define NE   800000
#define KDIM 128     // F_IN == HID == 128
#define OUTF 64

typedef __attribute__((ext_vector_type(2))) float v2f;
typedef __attribute__((ext_vector_type(8))) float v8f;

// ---------------- degree / normalization ----------------
__global__ void deg_init_kernel(float* deg, int n) {
    int i = blockIdx.x * blockDim.x + threadIdx.x;
    if (i < n) deg[i] = 1.0f;                       // self-loop contributes 1
}

__global__ void deg_accum_kernel(const int* __restrict__ dst, float* deg, int e) {
    int i = blockIdx.x * blockDim.x + threadIdx.x;
    if (i < e) atomicAdd(&deg[dst[i]], 1.0f);
}

__global__ void dinv_kernel(float* deg, int n) {
    int i = blockIdx.x * blockDim.x + threadIdx.x;
    if (i < n) deg[i] = rsqrtf(deg[i]);             // deg >= 1 always
}

// ---------------- fp32 WMMA GEMM: C[M x ncols] = A[M x 128] * B[128 x ncols] ----------------
// One wave per 16x16 output tile; K=128 -> 32 x v_wmma_f32_16x16x4_f32.
// A 16x4 f32 layout: lanes 0-15 row M=lane, K = k0+{0,1}; lanes 16-31 row M=lane-16, K = k0+{2,3}.
// B 4x16 f32 layout mirrored: lane = (k>=2)*16 + n, vgpr = k&1.
// C/D: VGPR j -> row j (lanes 0-15) / row j+8 (lanes 16-31), col = lane&15.
__global__ void gemm_wmma_f32_kernel(const float* __restrict__ A,
                                     const float* __restrict__ B,
                                     float* __restrict__ C, int ncols) {
    const int lane  = threadIdx.x & 31;
    const int tileM = blockIdx.x * 16;
    const int tileN = (blockIdx.y * blockDim.y + threadIdx.y) * 16;
    const int n16   = lane & 15;
    const int koff  = (lane >> 4) << 1;   // 0 for lanes 0-15, 2 for lanes 16-31

    const float* arow = A + (size_t)(tileM + n16) * KDIM;
    v8f acc = {};
#pragma unroll 8
    for (int k0 = 0; k0 < KDIM; k0 += 4) {
        v2f a = *(const v2f*)(arow + k0 + koff);            // global_load_b64
        v2f b;
        b.x = B[(size_t)(k0 + koff)     * ncols + tileN + n16];
        b.y = B[(size_t)(k0 + koff + 1) * ncols + tileN + n16];
        acc = __builtin_amdgcn_wmma_f32_16x16x4_f32(
            /*neg_a=*/false, a, /*neg_b=*/false, b,
            /*c_mod=*/(short)0, acc, /*reuse_a=*/false, /*reuse_b=*/false);
    }
    float* crow = C + (size_t)(tileM + ((lane >> 4) << 3)) * ncols + tileN + n16;
#pragma unroll
    for (int j = 0; j < 8; ++j) crow[(size_t)j * ncols] = acc[j];
}

// ---------------- layer-1 aggregation ----------------
// Init agg with the self-loop term: agg[i,:] = dinv[i]^2 * h_pre[i,:]
__global__ void selfloop128_kernel(const float* __restrict__ h,
                                   const float* __restrict__ dinv,
                                   float* __restrict__ agg, int total) {
    int idx = blockIdx.x * blockDim.x + threadIdx.x;
    if (idx < total) {
        float d = dinv[idx >> 7];
        agg[idx] = d * d * h[idx];
    }
}

// One wave per edge: float4 gather per lane (32*4 = 128 feats), atomic scatter.
__global__ void edge_scatter128_kernel(const float* __restrict__ h,
                                       const float* __restrict__ dinv,
                                       const int* __restrict__ src,
                                       const int* __restrict__ dst,
                                       float* __restrict__ agg, int e) {
    int eid = blockIdx.x * (blockDim.x >> 5) + (threadIdx.x >> 5);
    if (eid >= e) return;
    int lane = threadIdx.x & 31;
    int s = src[eid], d = dst[eid];
    float nrm = dinv[s] * dinv[d];
    float4 v = ((const float4*)(h + (size_t)s * 128))[lane];
    float* p = agg + (size_t)d * 128 + lane * 4;
    atomicAdd(p + 0, nrm * v.x);
    atomicAdd(p + 1, nrm * v.y);
    atomicAdd(p + 2, nrm * v.z);
    atomicAdd(p + 3, nrm * v.w);
}

// +b1, ReLU, per-node LayerNorm over 128 feats. One wave per node, 4 feats/lane.
__global__ void bias_relu_ln_kernel(float* __restrict__ h,
                                    const float* __restrict__ b1,
                                    const float* __restrict__ gamma,
                                    const float* __restrict__ beta, int n) {
    int node = blockIdx.x * (blockDim.x >> 5) + (threadIdx.x >> 5);
    if (node >= n) return;
    int lane = threadIdx.x & 31;
    float4 v  = ((float4*)(h + (size_t)node * 128))[lane];
    float4 bb = ((const float4*)b1)[lane];
    v.x = fmaxf(v.x + bb.x, 0.0f);
    v.y = fmaxf(v.y + bb.y, 0.0f);
    v.z = fmaxf(v.z + bb.z, 0.0f);
    v.w = fmaxf(v.w + bb.w, 0.0f);
    float s  = v.x + v.y + v.z + v.w;
    float s2 = v.x * v.x + v.y * v.y + v.z * v.z + v.w * v.w;
#pragma unroll
    for (int o = 16; o > 0; o >>= 1) {       // wave32 reduction
        s  += __shfl_xor(s,  o, 32);
        s2 += __shfl_xor(s2, o, 32);
    }
    float mean = s * (1.0f / 128.0f);
    float var  = s2 * (1.0f / 128.0f) - mean * mean;
    float inv  = rsqrtf(var + 1e-5f);
    float4 g  = ((const float4*)gamma)[lane];
    float4 be = ((const float4*)beta)[lane];
    v.x = (v.x - mean) * inv * g.x + be.x;
    v.y = (v.y - mean) * inv * g.y + be.y;
    v.z = (v.z - mean) * inv * g.z + be.z;
    v.w = (v.w - mean) * inv * g.w + be.w;
    ((float4*)(h + (size_t)node * 128))[lane] = v;
}

// ---------------- layer-2 aggregation (mu + logvar together) ----------------
// Init out with self-loop term and bias (bias is added after segment_sum).
__global__ void selfloop64_bias_kernel(const float* __restrict__ mup,
                                       const float* __restrict__ lvp,
                                       const float* __restrict__ dinv,
                                       const float* __restrict__ bmu,
                                       const float* __restrict__ blv,
                                       float* __restrict__ mu,
                                       float* __restrict__ lv, int total) {
    int idx = blockIdx.x * blockDim.x + threadIdx.x;
    if (idx < total) {
        int f = idx & 63;
        float d = dinv[idx >> 6];
        float d2 = d * d;
        mu[idx] = d2 * mup[idx] + bmu[f];
        lv[idx] = d2 * lvp[idx] + blv[f];
    }
}

__global__ void edge_scatter64_kernel(const float* __restrict__ mup,
                                      const float* __restrict__ lvp,
                                      const float* __restrict__ dinv,
                                      const int* __restrict__ src,
                                      const int* __restrict__ dst,
                                      float* __restrict__ mu,
                                      float* __restrict__ lv, int e) {
    int eid = blockIdx.x * (blockDim.x >> 5) + (threadIdx.x >> 5);
    if (eid >= e) return;
    int lane = threadIdx.x & 31;
    int s = src[eid], d = dst[eid];
    float nrm = dinv[s] * dinv[d];
    float2 vm = ((const float2*)(mup + (size_t)s * 64))[lane];
    float2 vl = ((const float2*)(lvp + (size_t)s * 64))[lane];
    float* pm = mu + (size_t)d * 64 + lane * 2;
    float* pl = lv + (size_t)d * 64 + lane * 2;
    atomicAdd(pm + 0, nrm * vm.x);
    atomicAdd(pm + 1, nrm * vm.y);
    atomicAdd(pl + 0, nrm * vl.x);
    atomicAdd(pl + 1, nrm * vl.y);
}

extern "C" void kernel_launch(void* const* d_in, const int* in_sizes, int n_in,
                              void* d_out, int out_size, void* d_ws, size_t ws_size,
                              hipStream_t stream) {
    const float* x     = (const float*)d_in[0];
    const int*   ei    = (const int*)d_in[1];     // edge_index [2, E]
    const float* W1    = (const float*)d_in[2];
    const float* b1    = (const float*)d_in[3];
    const float* gamma = (const float*)d_in[4];
    const float* beta  = (const float*)d_in[5];
    const float* Wmu   = (const float*)d_in[6];
    const float* bmu   = (const float*)d_in[7];
    const float* Wlv   = (const float*)d_in[8];
    const float* blv   = (const float*)d_in[9];

    const int* src = ei;
    const int* dst = ei + NE;

    float* mu = (float*)d_out;                    // [N, 64]
    float* lv = mu + (size_t)NN * OUTF;           // [N, 64]

    float* ws   = (float*)d_ws;
    float* dinv = ws;                             // N
    float* h1   = dinv + NN;                      // N*128  (X @ W1)
    float* agg  = h1 + (size_t)NN * KDIM;         // N*128  (aggregated -> h)
    float* mup  = agg + (size_t)NN * KDIM;        // N*64
    float* lvp  = mup + (size_t)NN * OUTF;        // N*64

    // 1) degree + rsqrt
    deg_init_kernel<<<(NN + 255) / 256, 256, 0, stream>>>(dinv, NN);
    deg_accum_kernel<<<(NE + 255) / 256, 256, 0, stream>>>(dst, dinv, NE);
    dinv_kernel<<<(NN + 255) / 256, 256, 0, stream>>>(dinv, NN);

    // 2) h1 = X @ W1   (50000x128 @ 128x128), fp32 WMMA
    gemm_wmma_f32_kernel<<<dim3(NN / 16, KDIM / 64), dim3(32, 4), 0, stream>>>(x, W1, h1, KDIM);

    // 3) aggregate layer 1
    selfloop128_kernel<<<(NN * KDIM) / 256, 256, 0, stream>>>(h1, dinv, agg, NN * KDIM);
    edge_scatter128_kernel<<<(NE + 7) / 8, 256, 0, stream>>>(h1, dinv, src, dst, agg, NE);

    // 4) +b1, ReLU, LayerNorm (in place -> h)
    bias_relu_ln_kernel<<<(NN + 7) / 8, 256, 0, stream>>>(agg, b1, gamma, beta, NN);

    // 5) mu_pre = h @ Wmu, lv_pre = h @ Wlv  (50000x128 @ 128x64), fp32 WMMA
    gemm_wmma_f32_kernel<<<dim3(NN / 16, OUTF / 64), dim3(32, 4), 0, stream>>>(agg, Wmu, mup, OUTF);
    gemm_wmma_f32_kernel<<<dim3(NN / 16, OUTF / 64), dim3(32, 4), 0, stream>>>(agg, Wlv, lvp, OUTF);

    // 6) aggregate layer 2 directly into d_out (self-loop + bias init, then edges)
    selfloop64_bias_kernel<<<(NN * OUTF) / 256, 256, 0, stream>>>(mup, lvp, dinv, bmu, blv,
                                                                  mu, lv, NN * OUTF);
    edge_scatter64_kernel<<<(NE + 7) / 8, 256, 0, stream>>>(mup, lvp, dinv, src, dst, mu, lv, NE);
}